// TTEmbeddings_85401129713992
// MI455X (gfx1250) — compile-verified
//
#include <hip/hip_runtime.h>
#include <hip/hip_bf16.h>

typedef __attribute__((ext_vector_type(2))) float v2f;
typedef __attribute__((ext_vector_type(8))) float v8f;

// ---------------------------------------------------------------------------
// Stage 1: precompute P23[pair] = g2[i2] (32x8x32) contracted with g3[i3]
// (32x8) over r2  ->  (r1=32) x (m2*8+m3 = 64) panel, for all 2500 pairs.
// Pure VALU; 0.33 GMAC total, negligible. core3 slice cached in LDS.
// ---------------------------------------------------------------------------
__global__ __launch_bounds__(256) void tt_pair_precompute(
    const float* __restrict__ core2,  // (50, 32, 8, 32)
    const float* __restrict__ core3,  // (50, 32, 8, 1)
    float* __restrict__ p23)          // (2500, 32, 64)
{
    const int pair = blockIdx.x;          // 0..2499
    const int i2 = pair / 50;
    const int i3 = pair - i2 * 50;

    __shared__ float g3s[256];            // (r2=32, m3=8)
    for (int t = threadIdx.x; t < 256; t += blockDim.x)
        g3s[t] = core3[i3 * 256 + t];
    __syncthreads();

    const float* __restrict__ g2 = core2 + (size_t)i2 * 8192;  // r1*256 + m2*32 + r2
    float* __restrict__ out = p23 + (size_t)pair * 2048;       // r1*64 + m2*8 + m3

    for (int o = threadIdx.x; o < 2048; o += blockDim.x) {
        const int r1 = o >> 6;
        const int m2 = (o >> 3) & 7;
        const int m3 = o & 7;
        const float* __restrict__ a = g2 + r1 * 256 + m2 * 32;
        float s = 0.f;
#pragma unroll
        for (int r2 = 0; r2 < 32; ++r2)
            s = fmaf(a[r2], g3s[r2 * 8 + m3], s);
        out[o] = s;
    }
}

// ---------------------------------------------------------------------------
// Stage 2: one wave32 per lookup.
//   out(8x64) = g1(8x32) @ P23(32x64)   via V_WMMA_F32_16X16X4_F32
// M padded 8->16, 4 N-tiles of 16, 8 K-steps of 4.  A fragments (= whole g1)
// loaded once per wave and reused across the 4 N-tiles; B streams from the
// L2-resident P23 table.  Outputs stream to HBM (the only large HBM traffic).
// ---------------------------------------------------------------------------
__global__ __launch_bounds__(256) void tt_gather_wmma(
    const int*   __restrict__ word_idx,   // 8192
    const int*   __restrict__ ctx_idx,    // 81920
    const float* __restrict__ w_core1,    // (50, 1, 8, 32)
    const float* __restrict__ c_core1,    // (50, 1, 8, 32)
    const float* __restrict__ p23_w,      // (2500, 32, 64)
    const float* __restrict__ p23_c,      // (2500, 32, 64)
    float* __restrict__ out)              // [8192*512 | 81920*512]
{
    const int NWORD = 8192;
    const int wave  = (int)((blockIdx.x * blockDim.x + threadIdx.x) >> 5);
    const int lane  = (int)(threadIdx.x & 31);
    if (wave >= NWORD + 81920) return;    // wave-uniform; full-wave exit only

    int idx;
    const float* core1;
    const float* p23;
    size_t obase;
    if (wave < NWORD) {
        idx   = word_idx[wave];
        core1 = w_core1;
        p23   = p23_w;
        obase = (size_t)wave * 512;
    } else {
        const int j = wave - NWORD;
        idx   = ctx_idx[j];
        core1 = c_core1;
        p23   = p23_c;
        obase = (size_t)NWORD * 512 + (size_t)j * 512;
    }

    const int i1 = idx / 2500;
    const int i2 = (idx / 50) % 50;
    const int i3 = idx % 50;

    const float* __restrict__ g1 = core1 + (size_t)i1 * 256;          // m1*32 + r1
    const float* __restrict__ P  = p23 + (size_t)(i2 * 50 + i3) * 2048; // r1*64 + n

    const int m     = lane & 15;          // row index within 16x16 tile
    const int khalf = (lane >> 4) << 1;   // lanes 0-15: K+0/K+1, lanes 16-31: K+2/K+3

    // A fragments: rows 0-7 = g1, rows 8-15 = zero padding.
    v2f a[8];
#pragma unroll
    for (int kk = 0; kk < 8; ++kk) {
        const int r1 = kk * 4 + khalf;
        float ax = 0.f, ay = 0.f;
        if (m < 8) {
            ax = g1[m * 32 + r1];
            ay = g1[m * 32 + r1 + 1];
        }
        a[kk].x = ax;
        a[kk].y = ay;
    }

    float* __restrict__ o = out + obase;
#pragma unroll
    for (int nt = 0; nt < 4; ++nt) {
        const int n = nt * 16 + (lane & 15);
        v8f acc = {};
#pragma unroll
        for (int kk = 0; kk < 8; ++kk) {
            const int r1 = kk * 4 + khalf;
            v2f b;
            b.x = P[r1 * 64 + n];
            b.y = P[r1 * 64 + 64 + n];
            acc = __builtin_amdgcn_wmma_f32_16x16x4_f32(
                /*neg_a=*/false, a[kk], /*neg_b=*/false, b,
                /*c_mod=*/(short)0, acc, /*reuse_a=*/false, /*reuse_b=*/false);
        }
        // D rows 0-7 live in lanes 0-15 (VGPR g = row g); rows 8-15 are padding.
        if (lane < 16) {
#pragma unroll
            for (int g = 0; g < 8; ++g)
                o[g * 64 + n] = acc[g];
        }
    }
}

extern "C" void kernel_launch(void* const* d_in, const int* in_sizes, int n_in,
                              void* d_out, int out_size, void* d_ws, size_t ws_size,
                              hipStream_t stream) {
    const int*   wi = (const int*)d_in[0];
    const int*   ci = (const int*)d_in[1];
    const float* w1 = (const float*)d_in[2];
    const float* w2 = (const float*)d_in[3];
    const float* w3 = (const float*)d_in[4];
    const float* c1 = (const float*)d_in[5];
    const float* c2 = (const float*)d_in[6];
    const float* c3 = (const float*)d_in[7];
    float* out = (float*)d_out;

    // Workspace: two P23 tables, 2500 * 2048 floats each (20.5 MB each).
    float* p23w = (float*)d_ws;
    float* p23c = p23w + (size_t)2500 * 2048;

    tt_pair_precompute<<<2500, 256, 0, stream>>>(w2, w3, p23w);
    tt_pair_precompute<<<2500, 256, 0, stream>>>(c2, c3, p23c);

    // 90112 lookups, 1 wave each, 8 waves per 256-thread block.
    tt_gather_wmma<<<(8192 + 81920) / 8, 256, 0, stream>>>(
        wi, ci, w1, c1, p23w, p23c, out);
}